// geo_GAT2_45045617000456
// MI455X (gfx1250) — compile-verified
//
#include <hip/hip_runtime.h>
#include <stdint.h>

#define N_NODES 50000
#define N_EDGES 800000
#define IN_C    256
#define NHID    32
#define HEADS   4
#define HID     128   // HEADS*NHID
#define OUT_C   64
#define SLOPE   0.2f

typedef float v2f __attribute__((ext_vector_type(2)));
typedef float v8f __attribute__((ext_vector_type(8)));

__device__ __forceinline__ float leaky_d(float x) { return x >= 0.f ? x : SLOPE * x; }

// monotone float<->uint encoding for atomicMax on floats
__device__ __forceinline__ unsigned fenc(float f) {
    unsigned u = __float_as_uint(f);
    return (u & 0x80000000u) ? ~u : (u | 0x80000000u);
}
__device__ __forceinline__ float fdec(unsigned u) {
    return (u & 0x80000000u) ? __uint_as_float(u ^ 0x80000000u) : __uint_as_float(~u);
}

// ---------------------------------------------------------------------------
// FP32 WMMA GEMM: C[M,Ncol] = op(A)[M,K] @ B[K,Ncol] (+ optional bias row)
// Block = 128 threads (4 waves). The 16-wide B column-tile (K x 16) is staged
// in LDS once per block; each wave then computes one 16x16 output tile with
// V_WMMA_F32_16X16X4_F32.
//   A-frag (16x4 f32): lane<16 -> M=lane; VGPR0=K+2g, VGPR1=K+2g+1 (g=lane>>4)
//   B-frag (4x16 f32): VGPRj, group g -> row K=j+2g, col=lane&15
//   C/D:               VGPR r, group g -> row M=r+8g,  col=lane&15
// ---------------------------------------------------------------------------
template <int APPLY_LEAKY, int HAS_BIAS>
__global__ void wmma_gemm_f32(const float* __restrict__ A, const float* __restrict__ B,
                              const float* __restrict__ bias, float* __restrict__ C,
                              int K, int Ncol, int rowTiles)
{
    __shared__ float Bs[IN_C * 16];   // max K = 256 -> 16 KB of 320 KB/WGP

    const int tid  = threadIdx.x;
    const int col0 = blockIdx.y * 16;

    // cooperative stage of B tile (K x 16) into LDS, coalesced
    for (int i = tid; i < K * 16; i += 128) {
        const int kk = i >> 4, cc = i & 15;
        Bs[i] = B[(size_t)kk * Ncol + col0 + cc];
    }
    __syncthreads();

    const int wave = tid >> 5;
    const int lane = tid & 31;
    const int rowT = blockIdx.x * 4 + wave;
    if (rowT >= rowTiles) return;     // wave-uniform exit; EXEC stays all-1s

    const int laneLo = lane & 15;
    const int grp    = lane >> 4;
    const int row0   = rowT * 16;

    const float* Arow = A + (size_t)(row0 + laneLo) * K + 2 * grp;  // 8B aligned
    const float* Bp   = Bs + 2 * grp * 16 + laneLo;

    v8f acc = {};
#pragma unroll 4
    for (int k = 0; k < K; k += 4) {
        v2f a = *(const v2f*)(Arow + k);          // global_load_b64
        if (APPLY_LEAKY) { a.x = leaky_d(a.x); a.y = leaky_d(a.y); }
        v2f b;
        b.x = Bp[k * 16];                          // ds_load
        b.y = Bp[k * 16 + 16];
        acc = __builtin_amdgcn_wmma_f32_16x16x4_f32(false, a, false, b,
                                                    (short)0, acc, false, false);
    }

    const float bb = HAS_BIAS ? bias[col0 + laneLo] : 0.f;
#pragma unroll
    for (int r = 0; r < 8; ++r) {
        const int row = row0 + r + 8 * grp;
        C[(size_t)row * Ncol + col0 + laneLo] = acc[r] + bb;
    }
}

// a_src[n,h] = sum_c H[n,h,c]*attS[h,c] ; a_dst likewise
template <int H, int C>
__global__ void node_attn(const float* __restrict__ Hf, const float* __restrict__ attS,
                          const float* __restrict__ attD,
                          float* __restrict__ aS, float* __restrict__ aD)
{
    const int gid = blockIdx.x * blockDim.x + threadIdx.x;
    if (gid >= N_NODES * H) return;
    const int node = gid / H, h = gid % H;
    const float* row = Hf + (size_t)node * (H * C) + h * C;
    float s = 0.f, d = 0.f;
#pragma unroll 8
    for (int c = 0; c < C; ++c) {
        const float v = row[c];
        s += v * attS[h * C + c];
        d += v * attD[h * C + c];
    }
    aS[gid] = s; aD[gid] = d;
}

__global__ void fill_f32(float* p, float v, int n) {
    const int i = blockIdx.x * blockDim.x + threadIdx.x;
    if (i < n) p[i] = v;
}
__global__ void fill_u32(unsigned* p, unsigned v, int n) {
    const int i = blockIdx.x * blockDim.x + threadIdx.x;
    if (i < n) p[i] = v;
}

// pass 1: alpha = leaky(aS[src]+aD[dst]); store; segment-max over dst
template <int LOG_H>
__global__ void edge_alpha_max(const int* __restrict__ ei,
                               const float* __restrict__ aS, const float* __restrict__ aD,
                               float* __restrict__ alphaBuf, unsigned* __restrict__ amax,
                               int M, int E)
{
    constexpr int H = 1 << LOG_H;
    const int gid = blockIdx.x * blockDim.x + threadIdx.x;
    if (gid >= (M << LOG_H)) return;
    const int e = gid >> LOG_H, h = gid & (H - 1);
    int s, d;
    if (e < E) { s = ei[e]; d = ei[E + e]; } else { s = d = e - E; }
    const float al = leaky_d(aS[(s << LOG_H) + h] + aD[(d << LOG_H) + h]);
    alphaBuf[gid] = al;
    atomicMax(&amax[(d << LOG_H) + h], fenc(al));
}

// pass 2: expa = exp(alpha - amax[dst]); store; segment-sum over dst
template <int LOG_H>
__global__ void edge_exp_sum(const int* __restrict__ ei,
                             float* __restrict__ alphaBuf,
                             const unsigned* __restrict__ amax, float* __restrict__ denom,
                             int M, int E)
{
    constexpr int H = 1 << LOG_H;
    const int gid = blockIdx.x * blockDim.x + threadIdx.x;
    if (gid >= (M << LOG_H)) return;
    const int e = gid >> LOG_H, h = gid & (H - 1);
    int d;
    if (e < E) { d = ei[E + e]; } else { d = e - E; }
    const float ex = __expf(alphaBuf[gid] - fdec(amax[(d << LOG_H) + h]));
    alphaBuf[gid] = ex;
    atomicAdd(&denom[(d << LOG_H) + h], ex);
}

// pass 3 (heavy): out[dst,c] += (expa/denom[dst]) * Hf[src,c]  (coalesced over c)
template <int LOG_H, int LOG_CT>
__global__ void edge_scatter(const int* __restrict__ ei,
                             const float* __restrict__ alphaBuf, const float* __restrict__ denom,
                             const float* __restrict__ Hf, float* __restrict__ out,
                             int M, int E)
{
    constexpr int CT = 1 << LOG_CT;
    const int gid = blockIdx.x * blockDim.x + threadIdx.x;
    if (gid >= (M << LOG_CT)) return;
    const int e = gid >> LOG_CT, c = gid & (CT - 1);
    const int h = c >> (LOG_CT - LOG_H);
    int s, d;
    if (e < E) { s = ei[e]; d = ei[E + e]; } else { s = d = e - E; }
    const float coeff = alphaBuf[(e << LOG_H) + h] / (denom[(d << LOG_H) + h] + 1e-16f);
    atomicAdd(&out[((size_t)d << LOG_CT) + c], coeff * Hf[((size_t)s << LOG_CT) + c]);
}

// h1 = out1 + b1 + xres   (in place into out1)
__global__ void combine1(float* __restrict__ out1, const float* __restrict__ b1,
                         const float* __restrict__ xres)
{
    const int gid = blockIdx.x * blockDim.x + threadIdx.x;
    if (gid >= N_NODES * HID) return;
    out1[gid] = out1[gid] + b1[gid & (HID - 1)] + xres[gid];
}

// z = leaky(out2 + b2 + hres)
__global__ void combine2(const float* __restrict__ out2, const float* __restrict__ b2,
                         const float* __restrict__ hres, float* __restrict__ z)
{
    const int gid = blockIdx.x * blockDim.x + threadIdx.x;
    if (gid >= N_NODES * OUT_C) return;
    z[gid] = leaky_d(out2[gid] + b2[gid & (OUT_C - 1)] + hres[gid]);
}

static inline size_t alignup(size_t v) { return (v + 255) & ~(size_t)255; }

extern "C" void kernel_launch(void* const* d_in, const int* in_sizes, int n_in,
                              void* d_out, int out_size, void* d_ws, size_t ws_size,
                              hipStream_t stream)
{
    const float* x     = (const float*)d_in[0];
    const int*   ei    = (const int*)d_in[1];     // [2,E] flat: src then dst
    const float* W1    = (const float*)d_in[2];
    const float* attS1 = (const float*)d_in[3];
    const float* attD1 = (const float*)d_in[4];
    const float* b1    = (const float*)d_in[5];
    const float* W2    = (const float*)d_in[6];
    const float* attS2 = (const float*)d_in[7];
    const float* attD2 = (const float*)d_in[8];
    const float* b2    = (const float*)d_in[9];
    const float* r1w   = (const float*)d_in[10];
    const float* r1b   = (const float*)d_in[11];
    const float* fcw   = (const float*)d_in[12];
    const float* fcb   = (const float*)d_in[13];
    float* z = (float*)d_out;

    const int M = N_EDGES + N_NODES;  // edges + self loops

    char* p = (char*)d_ws;
    float*    bufH     = (float*)p;    p += alignup((size_t)N_NODES * HID * 4);   // x@W1, later H2
    float*    bufXres  = (float*)p;    p += alignup((size_t)N_NODES * HID * 4);   // xres, later Hres
    float*    bufOut1  = (float*)p;    p += alignup((size_t)N_NODES * HID * 4);   // out1 -> h1
    float*    bufOut2  = (float*)p;    p += alignup((size_t)N_NODES * OUT_C * 4);
    float*    bufAlpha = (float*)p;    p += alignup((size_t)M * HEADS * 4);
    unsigned* bufAmax  = (unsigned*)p; p += alignup((size_t)N_NODES * HEADS * 4);
    float*    bufDen   = (float*)p;    p += alignup((size_t)N_NODES * HEADS * 4);
    float*    bufAS    = (float*)p;    p += alignup((size_t)N_NODES * HEADS * 4);
    float*    bufAD    = (float*)p;    p += alignup((size_t)N_NODES * HEADS * 4);

    const int TB = 256;
    const int rowTiles  = N_NODES / 16;              // 3125 exact
    const int rowBlocks = (rowTiles + 3) / 4;        // 4 waves (tiles) per block

    // ---- layer 1 GEMMs (WMMA fp32, LDS-staged B) ----
    wmma_gemm_f32<0, 0><<<dim3(rowBlocks, HID / 16), 128, 0, stream>>>(x, W1, nullptr, bufH, IN_C, HID, rowTiles);
    wmma_gemm_f32<0, 1><<<dim3(rowBlocks, HID / 16), 128, 0, stream>>>(x, r1w, r1b, bufXres, IN_C, HID, rowTiles);

    // ---- layer 1 attention ----
    node_attn<HEADS, NHID><<<(N_NODES * HEADS + TB - 1) / TB, TB, 0, stream>>>(bufH, attS1, attD1, bufAS, bufAD);
    fill_u32<<<(N_NODES * HEADS + TB - 1) / TB, TB, 0, stream>>>(bufAmax, 0u, N_NODES * HEADS);
    fill_f32<<<(N_NODES * HEADS + TB - 1) / TB, TB, 0, stream>>>(bufDen, 0.f, N_NODES * HEADS);
    fill_f32<<<(N_NODES * HID + TB - 1) / TB, TB, 0, stream>>>(bufOut1, 0.f, N_NODES * HID);

    edge_alpha_max<2><<<(M * HEADS + TB - 1) / TB, TB, 0, stream>>>(ei, bufAS, bufAD, bufAlpha, bufAmax, M, N_EDGES);
    edge_exp_sum<2>  <<<(M * HEADS + TB - 1) / TB, TB, 0, stream>>>(ei, bufAlpha, bufAmax, bufDen, M, N_EDGES);
    edge_scatter<2, 7><<<((size_t)M * HID + TB - 1) / TB, TB, 0, stream>>>(ei, bufAlpha, bufDen, bufH, bufOut1, M, N_EDGES);

    // h1 = out1 + b1 + xres (in place)
    combine1<<<(N_NODES * HID + TB - 1) / TB, TB, 0, stream>>>(bufOut1, b1, bufXres);

    // ---- layer 2 GEMMs: Hres = h1@fcw+fcb ; H2 = leaky(h1)@W2 ----
    wmma_gemm_f32<0, 1><<<dim3(rowBlocks, OUT_C / 16), 128, 0, stream>>>(bufOut1, fcw, fcb, bufXres, HID, OUT_C, rowTiles);
    wmma_gemm_f32<1, 0><<<dim3(rowBlocks, OUT_C / 16), 128, 0, stream>>>(bufOut1, W2, nullptr, bufH, HID, OUT_C, rowTiles);

    // ---- layer 2 attention (1 head, 64 ch) ----
    node_attn<1, OUT_C><<<(N_NODES + TB - 1) / TB, TB, 0, stream>>>(bufH, attS2, attD2, bufAS, bufAD);
    fill_u32<<<(N_NODES + TB - 1) / TB, TB, 0, stream>>>(bufAmax, 0u, N_NODES);
    fill_f32<<<(N_NODES + TB - 1) / TB, TB, 0, stream>>>(bufDen, 0.f, N_NODES);
    fill_f32<<<(N_NODES * OUT_C + TB - 1) / TB, TB, 0, stream>>>(bufOut2, 0.f, N_NODES * OUT_C);

    edge_alpha_max<0><<<(M + TB - 1) / TB, TB, 0, stream>>>(ei, bufAS, bufAD, bufAlpha, bufAmax, M, N_EDGES);
    edge_exp_sum<0>  <<<(M + TB - 1) / TB, TB, 0, stream>>>(ei, bufAlpha, bufAmax, bufDen, M, N_EDGES);
    edge_scatter<0, 6><<<((size_t)M * OUT_C + TB - 1) / TB, TB, 0, stream>>>(ei, bufAlpha, bufDen, bufH, bufOut2, M, N_EDGES);

    // z = leaky(out2 + b2 + Hres)
    combine2<<<(N_NODES * OUT_C + TB - 1) / TB, TB, 0, stream>>>(bufOut2, b2, bufXres, z);

    (void)in_sizes; (void)n_in; (void)out_size; (void)ws_size;
}